// TotalLoss_vis_atten_thm_5_23991687315992
// MI455X (gfx1250) — compile-verified
//
#include <hip/hip_runtime.h>
#include <hip/hip_bf16.h>
#include <math.h>

#define B_   8
#define HO_  152
#define WO_  272
#define MO_  128
#define NID_ 537
#define HW_  (HO_ * WO_)     // 41344
#define PADN 576             // 9 strips of 64 = 36 tiles of 16; zero-padded N
#define NROW (B_ * MO_)      // 1024

typedef __attribute__((ext_vector_type(2))) float v2f;
typedef __attribute__((ext_vector_type(8))) float v8f;

__device__ __forceinline__ float wave_sum(float v) {
#pragma unroll
    for (int off = 16; off > 0; off >>= 1) v += __shfl_xor(v, off, 32);
    return v;
}
__device__ __forceinline__ float wave_max(float v) {
#pragma unroll
    for (int off = 16; off > 0; off >>= 1) v = fmaxf(v, __shfl_xor(v, off, 32));
    return v;
}
__device__ __forceinline__ float emb_scale() {
    // sqrt(2) * log(NID - 1)
    return 1.41421356237f * logf((float)(NID_ - 1));
}

// ---------------- zero small accumulator region ----------------
__global__ void k_init(float* p, int n) {
    int i = blockIdx.x * blockDim.x + threadIdx.x;
    if (i < n) p[i] = 0.0f;
}

// ---------------- pad classifier weights: Wpad[n,k] = (n<NID) ? W[n,k] : 0 ----------------
__global__ void k_padw(const float* __restrict__ W, float* __restrict__ Wp, int K) {
    int i = blockIdx.x * blockDim.x + threadIdx.x;
    int total = PADN * K;
    if (i < total) {
        int n = i / K;
        Wp[i] = (n < NID_) ? W[i] : 0.0f;
    }
}

// ---------------- focal loss partial sums per batch ----------------
// grid: (chunks, B), block 256; each block handles 2048 elements of one batch.
__global__ void k_focal(const float* __restrict__ cls, const float* __restrict__ hm,
                        float* __restrict__ facc) {
    const int b = blockIdx.y;
    const int base = blockIdx.x * 2048;
    const int tid = threadIdx.x;
    __shared__ float s0[256], s1[256], s2[256];
    float np = 0.f, pl = 0.f, nl = 0.f;
#pragma unroll
    for (int i = 0; i < 8; ++i) {
        int e = base + tid + i * 256;
        if (e < HW_) {
            float x = cls[(size_t)b * HW_ + e];
            float g = hm [(size_t)b * HW_ + e];
            float p = 1.f / (1.f + expf(-x));
            p = fminf(fmaxf(p, 1e-4f), 1.f - 1e-4f);
            float pos  = (g == 1.0f) ? 1.f : 0.f;
            float om   = 1.f - g;
            float negw = om * om; negw *= negw;
            float omp  = 1.f - p;
            pl += logf(p) * omp * omp * pos;
            nl += logf(1.f - p) * p * p * negw * (1.f - pos);
            np += pos;
        }
    }
    s0[tid] = np; s1[tid] = pl; s2[tid] = nl;
    __syncthreads();
    for (int st = 128; st > 0; st >>= 1) {
        if (tid < st) { s0[tid] += s0[tid + st]; s1[tid] += s1[tid + st]; s2[tid] += s2[tid + st]; }
        __syncthreads();
    }
    if (tid == 0) {
        atomicAdd(&facc[3 * b + 0], s0[0]);
        atomicAdd(&facc[3 * b + 1], s1[0]);
        atomicAdd(&facc[3 * b + 2], s2[0]);
    }
}

// ---------------- gather + normalize embedding rows, reg/vis L1 ----------------
__device__ __forceinline__ void norm_row(const float* __restrict__ feat, int C,
                                         float* __restrict__ dst, int r, int b,
                                         int idx, int lane) {
    float tmp[8];
    float ss = 0.f;
    const int nIt = C / 32;
    for (int i = 0; i < nIt; ++i) {
        int c = lane + 32 * i;
        float x = feat[((size_t)(b * C + c)) * HW_ + idx];
        tmp[i] = x;
        ss += x * x;
    }
    ss = wave_sum(ss);
    float s = emb_scale() / fmaxf(sqrtf(ss), 1e-12f);
    for (int i = 0; i < nIt; ++i) {
        int c = lane + 32 * i;
        dst[(size_t)r * C + c] = tmp[i] * s;
    }
}

// grid: 1024 blocks, 32 threads (one wave per (b,m) row)
__global__ void k_gather(const float* __restrict__ regs, const float* __restrict__ viss,
                         const float* __restrict__ f64,  const float* __restrict__ f128,
                         const float* __restrict__ f256, const float* __restrict__ fend,
                         const float* __restrict__ wh,   const float* __restrict__ v,
                         const int* __restrict__ ind,    const int* __restrict__ msk,
                         float* __restrict__ regacc,
                         float* __restrict__ Fn64, float* __restrict__ Fn128,
                         float* __restrict__ Fn256, float* __restrict__ FnEnd) {
    const int r = blockIdx.x;
    const int b = r / MO_;
    const int lane = threadIdx.x;
    const int idx = ind[r];
    const float mk = (float)msk[r];

    // reg L1 over 8 channels (lanes 0..7)
    float rv = 0.f;
    if (lane < 8) {
        float pr = regs[((size_t)(b * 8 + lane)) * HW_ + idx];
        rv = fabsf(pr - wh[(size_t)r * 8 + lane]);
    }
    rv = wave_sum(rv);
    if (lane == 0) {
        atomicAdd(&regacc[b], mk * rv);
        float pv = 1.f / (1.f + expf(-viss[(size_t)b * HW_ + idx]));
        pv = fminf(fmaxf(pv, 1e-4f), 1.f - 1e-4f);
        atomicAdd(&regacc[B_ + b], mk * fabsf(pv - v[r]));
        atomicAdd(&regacc[2 * B_ + b], mk);
    }
    norm_row(f64,  64,  Fn64,  r, b, idx, lane);
    norm_row(f128, 128, Fn128, r, b, idx, lane);
    norm_row(f256, 256, Fn256, r, b, idx, lane);
    norm_row(fend, 128, FnEnd, r, b, idx, lane);
}

// ---------------- WMMA f32 GEMM: logits = Fn(1024xK) * Wpad^T(KxPADN) + bias ----------------
// Each wave computes a 16(M) x 64(N) strip = 4 tiles with independent accumulators,
// reusing the A operand across the 4 WMMAs per K-step.
// grid: (PADN/64, 1024/16), block 32. EXEC all ones at every WMMA.
__global__ void k_gemm(const float* __restrict__ A, const float* __restrict__ Wp,
                       const float* __restrict__ bias, float* __restrict__ out, int K) {
    const int lane = threadIdx.x;
    const int colBase = blockIdx.x * 64;
    const int rowBase = blockIdx.y * 16;
    const int mloc = lane & 15;
    const int half = lane >> 4;
    const int koff = half * 2;

    v8f c0 = {}, c1 = {}, c2 = {}, c3 = {};
    const float* ap  = A  + (size_t)(rowBase + mloc) * K + koff;
    const float* bp0 = Wp + (size_t)(colBase +  0 + mloc) * K + koff;
    const float* bp1 = Wp + (size_t)(colBase + 16 + mloc) * K + koff;
    const float* bp2 = Wp + (size_t)(colBase + 32 + mloc) * K + koff;
    const float* bp3 = Wp + (size_t)(colBase + 48 + mloc) * K + koff;

    for (int k0 = 0; k0 < K; k0 += 4) {
        v2f a, b0, b1, b2, b3;
        a.x  = ap[k0];  a.y  = ap[k0 + 1];
        b0.x = bp0[k0]; b0.y = bp0[k0 + 1];
        b1.x = bp1[k0]; b1.y = bp1[k0 + 1];
        b2.x = bp2[k0]; b2.y = bp2[k0 + 1];
        b3.x = bp3[k0]; b3.y = bp3[k0 + 1];
        c0 = __builtin_amdgcn_wmma_f32_16x16x4_f32(false, a, false, b0, (short)0, c0, false, false);
        c1 = __builtin_amdgcn_wmma_f32_16x16x4_f32(false, a, false, b1, (short)0, c1, false, false);
        c2 = __builtin_amdgcn_wmma_f32_16x16x4_f32(false, a, false, b2, (short)0, c2, false, false);
        c3 = __builtin_amdgcn_wmma_f32_16x16x4_f32(false, a, false, b3, (short)0, c3, false, false);
    }

#pragma unroll
    for (int t = 0; t < 4; ++t) {
        v8f ct = (t == 0) ? c0 : (t == 1) ? c1 : (t == 2) ? c2 : c3;
        int col = colBase + 16 * t + mloc;
        float bb = (col < NID_) ? bias[col] : 0.f;
#pragma unroll
        for (int rr = 0; rr < 8; ++rr) {
            int row = rowBase + rr + half * 8;
            out[(size_t)row * PADN + col] = ct[rr] + bb;
        }
    }
}

// ---------------- per-row log-softmax pick, accumulate per-batch ID sum ----------------
// grid: 1024 blocks, 32 threads
__global__ void k_rowpick(const float* __restrict__ logits, const int* __restrict__ ids,
                          const int* __restrict__ msk, const float* __restrict__ v,
                          float* __restrict__ idacc) {
    const int r = blockIdx.x;
    const int b = r / MO_;
    const int lane = threadIdx.x;
    const float* row = logits + (size_t)r * PADN;
    float mx = -3.4e38f;
    for (int nI = lane; nI < NID_; nI += 32) mx = fmaxf(mx, row[nI]);
    mx = wave_max(mx);
    float se = 0.f;
    for (int nI = lane; nI < NID_; nI += 32) se += expf(row[nI] - mx);
    se = wave_sum(se);
    if (lane == 0) {
        float picked = row[ids[r]] - mx - logf(se);
        atomicAdd(&idacc[b], (float)msk[r] * v[r] * picked);
    }
}

// ---------------- finalize four scalars ----------------
__global__ void k_final(const float* __restrict__ facc, const float* __restrict__ regacc,
                        const float* __restrict__ idacc, float* __restrict__ out) {
    if (threadIdx.x == 0) {
        float cs = 0.f, rs = 0.f, vs = 0.f, is = 0.f;
        for (int b = 0; b < B_; ++b) {
            float npos = facc[3 * b], pl = facc[3 * b + 1], nl = facc[3 * b + 2];
            cs += (npos > 0.f) ? (-(pl + nl) / fmaxf(npos, 1.f)) : (-nl);
            float cnt = regacc[2 * B_ + b];
            rs += regacc[b] / (cnt * 8.f + 1e-4f);
            vs += regacc[B_ + b] / (cnt * 1.f + 1e-4f);
            is += -idacc[b] / fmaxf(cnt, 1.f);
        }
        out[0] = cs / B_;
        out[1] = rs / B_;
        out[2] = vs / B_;
        out[3] = is / B_;
    }
}

extern "C" void kernel_launch(void* const* d_in, const int* in_sizes, int n_in,
                              void* d_out, int out_size, void* d_ws, size_t ws_size,
                              hipStream_t stream) {
    const float* cls   = (const float*)d_in[0];
    const float* regs  = (const float*)d_in[1];
    const float* viss  = (const float*)d_in[2];
    const float* f64   = (const float*)d_in[3];
    const float* f128  = (const float*)d_in[4];
    const float* f256  = (const float*)d_in[5];
    const float* fend  = (const float*)d_in[6];
    const float* hm    = (const float*)d_in[7];
    const float* wh    = (const float*)d_in[8];
    const float* v     = (const float*)d_in[9];
    const float* W64   = (const float*)d_in[10];
    const float* b64   = (const float*)d_in[11];
    const float* W128  = (const float*)d_in[12];
    const float* b128  = (const float*)d_in[13];
    const float* W256  = (const float*)d_in[14];
    const float* b256  = (const float*)d_in[15];
    const float* Wend  = (const float*)d_in[16];
    const float* bend  = (const float*)d_in[17];
    const int*   ind   = (const int*)d_in[20];
    const int*   rmask = (const int*)d_in[21];
    const int*   ids   = (const int*)d_in[22];
    float* out = (float*)d_out;

    // workspace layout (floats)
    float* ws     = (float*)d_ws;
    float* facc   = ws;                    // 3*B
    float* regacc = facc + 3 * B_;         // 3*B  (reg_num, vis_num, cnt)
    float* idacc  = regacc + 3 * B_;       // B
    float* Fn64   = idacc + B_;            // 1024*64
    float* Fn128  = Fn64  + (size_t)NROW * 64;
    float* Fn256  = Fn128 + (size_t)NROW * 128;
    float* FnEnd  = Fn256 + (size_t)NROW * 256;
    float* Wp64   = FnEnd + (size_t)NROW * 128;   // PADN*64 zero-padded weights
    float* Wp128  = Wp64  + (size_t)PADN * 64;
    float* Wp256  = Wp128 + (size_t)PADN * 128;
    float* WpEnd  = Wp256 + (size_t)PADN * 256;
    float* logits = WpEnd + (size_t)PADN * 128;   // 1024*PADN, reused across heads

    const int naccum = 3 * B_ + 3 * B_ + B_;
    k_init<<<1, 64, 0, stream>>>(ws, naccum);

    k_padw<<<(PADN * 64  + 255) / 256, 256, 0, stream>>>(W64,  Wp64,  64);
    k_padw<<<(PADN * 128 + 255) / 256, 256, 0, stream>>>(W128, Wp128, 128);
    k_padw<<<(PADN * 256 + 255) / 256, 256, 0, stream>>>(W256, Wp256, 256);
    k_padw<<<(PADN * 128 + 255) / 256, 256, 0, stream>>>(Wend, WpEnd, 128);

    dim3 fgrid((HW_ + 2047) / 2048, B_);
    k_focal<<<fgrid, 256, 0, stream>>>(cls, hm, facc);

    k_gather<<<NROW, 32, 0, stream>>>(regs, viss, f64, f128, f256, fend,
                                      wh, v, ind, rmask, regacc,
                                      Fn64, Fn128, Fn256, FnEnd);

    dim3 ggrid(PADN / 64, NROW / 16);   // 9 x 64 strips

    k_gemm<<<ggrid, 32, 0, stream>>>(Fn64, Wp64, b64, logits, 64);
    k_rowpick<<<NROW, 32, 0, stream>>>(logits, ids, rmask, v, idacc);

    k_gemm<<<ggrid, 32, 0, stream>>>(Fn128, Wp128, b128, logits, 128);
    k_rowpick<<<NROW, 32, 0, stream>>>(logits, ids, rmask, v, idacc);

    k_gemm<<<ggrid, 32, 0, stream>>>(Fn256, Wp256, b256, logits, 256);
    k_rowpick<<<NROW, 32, 0, stream>>>(logits, ids, rmask, v, idacc);

    k_gemm<<<ggrid, 32, 0, stream>>>(FnEnd, WpEnd, bend, logits, 128);
    k_rowpick<<<NROW, 32, 0, stream>>>(logits, ids, rmask, v, idacc);

    k_final<<<1, 32, 0, stream>>>(facc, regacc, idacc, out);
}